// TrajsFeatures_64080912056375
// MI455X (gfx1250) — compile-verified
//
#include <hip/hip_runtime.h>
#include <float.h>

typedef __attribute__((ext_vector_type(2))) float v2f;
typedef __attribute__((ext_vector_type(8))) float v8f;

#define TRAJ 128

__device__ __forceinline__ float waveAllSum(float v) {
#pragma unroll
  for (int off = 16; off > 0; off >>= 1) v += __shfl_xor(v, off, 32);
  return v;
}
__device__ __forceinline__ float waveAllMax(float v) {
#pragma unroll
  for (int off = 16; off > 0; off >>= 1) v = fmaxf(v, __shfl_xor(v, off, 32));
  return v;
}

// One wave (32 threads) per trajectory of 128 nodes. Lane l owns nodes 4l..4l+3.
__global__ __launch_bounds__(32)
void traj_node_kernel(const float* __restrict__ P,
                      float* __restrict__ X,       // [N,4]
                      float* __restrict__ scales,  // [G,2]
                      float* __restrict__ u_out,   // [G,3]
                      float* __restrict__ sf_ws,   // [G]
                      float* __restrict__ gmax_ws, // [G]
                      float* __restrict__ fv_ws)   // [G]
{
  const int g = blockIdx.x;
  const int lane = threadIdx.x;
  const int nb = g * TRAJ + lane * 4;
  const float invL = 1.0f / 128.0f;

  __shared__ float sBuf[256];     // [0..127] dr_norm, [128..255] dr_norm^2
  __shared__ float Xs[TRAJ * 4];  // staged X rows

  // ---- load 4 nodes (12 floats) as 3 aligned float4 ----
  const float4* Pv = (const float4*)(P + (size_t)nb * 3);
  float4 A4 = Pv[0], B4 = Pv[1], C4 = Pv[2];
  float px0 = A4.x, py0 = A4.y, pz0 = A4.z;
  float px1 = A4.w, py1 = B4.x, pz1 = B4.y;
  float px2 = B4.z, py2 = B4.w, pz2 = C4.x;
  float px3 = C4.y, py3 = C4.z, pz3 = C4.w;

  // neighbor for element 4l+3 comes from lane l+1's element 0
  float nx = __shfl_down(px0, 1, 32);
  float ny = __shfl_down(py0, 1, 32);
  float nz = __shfl_down(pz0, 1, 32);

  // ---- per-graph stats: mean, mean^2 -> P_STD ----
  float Sx  = waveAllSum(px0 + px1 + px2 + px3);
  float Sy  = waveAllSum(py0 + py1 + py2 + py3);
  float Sz  = waveAllSum(pz0 + pz1 + pz2 + pz3);
  float Sxx = waveAllSum(px0*px0 + px1*px1 + px2*px2 + px3*px3);
  float Syy = waveAllSum(py0*py0 + py1*py1 + py2*py2 + py3*py3);
  float Szz = waveAllSum(pz0*pz0 + pz1*pz1 + pz2*pz2 + pz3*pz3);
  float mx = Sx * invL, my = Sy * invL, mz = Sz * invL;
  float stdx = sqrtf(Sxx * invL - mx * mx);
  float stdy = sqrtf(Syy * invL - my * my);
  float stdz = sqrtf(Szz * invL - mz * mz);
  float PSTD = sqrtf(stdx*stdx + stdy*stdy + stdz*stdz);
  float sf = PSTD;

  // ---- raw dr (fill_last = 0) ----
  float dx0 = px1 - px0, dy0 = py1 - py0, dz0 = pz1 - pz0;
  float dx1 = px2 - px1, dy1 = py2 - py1, dz1 = pz2 - pz1;
  float dx2 = px3 - px2, dy2 = py3 - py2, dz2 = pz3 - pz2;
  float dx3 = (lane == 31) ? 0.0f : (nx - px3);
  float dy3 = (lane == 31) ? 0.0f : (ny - py3);
  float dz3 = (lane == 31) ? 0.0f : (nz - pz3);

  // ---- u = mean raw jump, normalized ----
  float Sdx = waveAllSum(dx0 + dx1 + dx2 + dx3);
  float Sdy = waveAllSum(dy0 + dy1 + dy2 + dy3);
  float Sdz = waveAllSum(dz0 + dz1 + dz2 + dz3);
  float ux = Sdx * invL, uy = Sdy * invL, uz = Sdz * invL;
  float un = sqrtf(1e-5f + ux*ux + uy*uy + uz*uz);
  ux /= un; uy /= un; uz /= un;

  // ---- dr_ = dr/sf ; dr_norm = sqrt(1e-5 + |dr_|^2) ----
  float q0x = dx0 / sf, q0y = dy0 / sf, q0z = dz0 / sf;
  float q1x = dx1 / sf, q1y = dy1 / sf, q1z = dz1 / sf;
  float q2x = dx2 / sf, q2y = dy2 / sf, q2z = dz2 / sf;
  float q3x = dx3 / sf, q3y = dy3 / sf, q3z = dz3 / sf;
  float xn0 = sqrtf(1e-5f + q0x*q0x + q0y*q0y + q0z*q0z);
  float xn1 = sqrtf(1e-5f + q1x*q1x + q1y*q1y + q1z*q1z);
  float xn2 = sqrtf(1e-5f + q2x*q2x + q2y*q2y + q2z*q2z);
  float xn3 = sqrtf(1e-5f + q3x*q3x + q3y*q3y + q3z*q3z);

  // ---- local inclusive cummax (element order) ----
  float r0 = xn0;
  float r1 = fmaxf(r0, xn1);
  float r2 = fmaxf(r1, xn2);
  float r3 = fmaxf(r2, xn3);
  float incl = r3;
#pragma unroll
  for (int off = 1; off < 32; off <<= 1) {
    float nv = __shfl_up(incl, off, 32);
    incl = (lane >= (unsigned)off) ? fmaxf(incl, nv) : incl;
  }
  float excl = __shfl_up(incl, 1, 32);
  excl = (lane == 0) ? -FLT_MAX : excl;
  float cm0 = fmaxf(excl, r0), cm1 = fmaxf(excl, r1);
  float cm2 = fmaxf(excl, r2), cm3 = fmaxf(excl, r3);
  float gmax = waveAllMax(r3);
  float fv = __shfl(xn0, 0, 32);  // dr_norm at first node of this graph

  // ---- stage scan inputs: rows 0-7 = dr_norm segments, rows 8-15 = dr_norm^2 ----
  sBuf[lane*4 + 0] = xn0; sBuf[lane*4 + 1] = xn1;
  sBuf[lane*4 + 2] = xn2; sBuf[lane*4 + 3] = xn3;
  sBuf[128 + lane*4 + 0] = xn0*xn0; sBuf[128 + lane*4 + 1] = xn1*xn1;
  sBuf[128 + lane*4 + 2] = xn2*xn2; sBuf[128 + lane*4 + 3] = xn3*xn3;
  __syncthreads();

  // ---- WMMA: S(16x16) = R(16x16) x U(16x16), U[p][n] = (p<=n) ----
  // A 16x4 layout: lane m=lane&15; lower half K=kk,kk+1; upper half K=kk+2,kk+3.
  // B 4x16 layout: VGPR0 -> row kk (lower) / kk+2 (upper); VGPR1 -> +1.
  const int m = lane & 15;
  const int hi = (lane >> 4) & 1;
  const int rowOff = ((m & 7) * 16) + ((m >> 3) << 7);
  v8f acc = {};
#pragma unroll
  for (int k = 0; k < 4; k++) {
    int kk = k * 4 + hi * 2;
    v2f a; a.x = sBuf[rowOff + kk]; a.y = sBuf[rowOff + kk + 1];
    v2f b; b.x = (kk     <= m) ? 1.0f : 0.0f;
           b.y = (kk + 1 <= m) ? 1.0f : 0.0f;
    acc = __builtin_amdgcn_wmma_f32_16x16x4_f32(false, a, false, b,
                                                (short)0, acc, false, false);
  }
  // Result: VGPR v, lanes 0-15 -> cum(dr_norm) seg v pos lane;
  //         lanes 16-31 -> cum(dr_norm^2) seg v pos lane-16.
  // Segment totals live at pos 15 (lane 15 / lane 31 per half).
  const int srcLane = (lane & 16) | 15;
  float t[8];
#pragma unroll
  for (int v = 0; v < 8; v++) t[v] = __shfl(acc[v], srcLane, 32);
  const int comp = 1 + hi;  // 1: cum_dist, 2: cum_msd
  float o = 0.0f;
#pragma unroll
  for (int v = 0; v < 8; v++) {
    Xs[(v * 16 + m) * 4 + comp] = acc[v] + o;
    o += t[v];
  }

  // element-layout features: time_norm and local cummax (pre-finalize)
#pragma unroll
  for (int j = 0; j < 4; j++) {
    int idx = lane * 4 + j;
    Xs[idx * 4 + 0] = (float)(idx + 1) * invL;
  }
  Xs[(lane*4 + 0) * 4 + 3] = cm0;
  Xs[(lane*4 + 1) * 4 + 3] = cm1;
  Xs[(lane*4 + 2) * 4 + 3] = cm2;
  Xs[(lane*4 + 3) * 4 + 3] = cm3;
  __syncthreads();

  // coalesced float4 stores of X rows
  float4* Xv = (float4*)(X + (size_t)(g * TRAJ) * 4);
  const float4* Xsv = (const float4*)Xs;
#pragma unroll
  for (int j = 0; j < 4; j++) Xv[lane * 4 + j] = Xsv[lane * 4 + j];

  if (lane == 0) {
    sf_ws[g]   = sf;
    gmax_ws[g] = gmax;
    fv_ws[g]   = fv;
    scales[g * 2 + 0] = PSTD;
    scales[g * 2 + 1] = 128.0f;
    u_out[g * 3 + 0] = ux;
    u_out[g * 3 + 1] = uy;
    u_out[g * 3 + 2] = uz;
  }
}

// Exclusive running-max over per-graph maxima (single block).
__global__ __launch_bounds__(1024)
void scan_max_kernel(const float* __restrict__ gmax,
                     float* __restrict__ pm, int G, int chunk)
{
  __shared__ float buf[1024];
  int t = threadIdx.x;
  int lo = t * chunk;
  int hiI = lo + chunk; if (hiI > G) hiI = G;
  float mv = -FLT_MAX;
  for (int i = lo; i < hiI; i++) mv = fmaxf(mv, gmax[i]);
  buf[t] = mv;
  __syncthreads();
  for (int off = 1; off < 1024; off <<= 1) {
    float nv = (t >= off) ? buf[t - off] : -FLT_MAX;
    __syncthreads();
    buf[t] = fmaxf(buf[t], nv);
    __syncthreads();
  }
  float run = (t == 0) ? -FLT_MAX : buf[t - 1];
  for (int i = lo; i < hiI; i++) {
    pm[i] = run;
    run = fmaxf(run, gmax[i]);
  }
}

// cmx[i] = max(local_cummax, prefix_max_of_prior_graphs) + dr_norm_first[g]
__global__ void finalize_cmx_kernel(float* __restrict__ X,
                                    const float* __restrict__ pm,
                                    const float* __restrict__ fv, int N)
{
  int i = blockIdx.x * blockDim.x + threadIdx.x;
  if (i < N) {
    int g = i >> 7;
    X[(size_t)i * 4 + 3] = fmaxf(X[(size_t)i * 4 + 3], pm[g]) + fv[g];
  }
}

__device__ __forceinline__ void loadP3(const float* __restrict__ P, int i,
                                       float& x, float& y, float& z) {
  const float* p = P + (size_t)i * 3;
  x = p[0]; y = p[1]; z = p[2];
}

__global__ void traj_edge_kernel(const float* __restrict__ P,
                                 const int* __restrict__ row,
                                 const int* __restrict__ col,
                                 const float* __restrict__ X,
                                 const float* __restrict__ sf_ws,
                                 float* __restrict__ Eo, int NE)
{
  int e = blockIdx.x * blockDim.x + threadIdx.x;
  if (e >= NE) return;
  int r = row[e], c = col[e];
  int gr = r >> 7, gc = c >> 7;
  int pr = r & 127, pc = c & 127;

  float4 xr = *(const float4*)(X + (size_t)r * 4);
  float4 xc = *(const float4*)(X + (size_t)c * 4);
  float sfr = sf_ws[gr], sfc = sf_ws[gc];

  float rx, ry, rz, cx, cy, cz;
  loadP3(P, r, rx, ry, rz);
  loadP3(P, c, cx, cy, cz);

  int rn = (pr == 127) ? r : r + 1;
  int cn = (pc == 127) ? c : c + 1;
  float rnx, rny, rnz, cnx, cny, cnz;
  loadP3(P, rn, rnx, rny, rnz);
  loadP3(P, cn, cnx, cny, cnz);

  float drx = ((pr == 127) ? 0.0f : (rnx - rx)) / sfr;
  float dry = ((pr == 127) ? 0.0f : (rny - ry)) / sfr;
  float drz = ((pr == 127) ? 0.0f : (rnz - rz)) / sfr;
  float dcx = ((pc == 127) ? 0.0f : (cnx - cx)) / sfc;
  float dcy = ((pc == 127) ? 0.0f : (cny - cy)) / sfc;
  float dcz = ((pc == 127) ? 0.0f : (cnz - cz)) / sfc;

  float pdx = rx / sfr - cx / sfc;
  float pdy = ry / sfr - cy / sfc;
  float pdz = rz / sfr - cz / sfc;

  float d    = sqrtf(pdx*pdx + pdy*pdy + pdz*pdz);
  float corr = drx*dcx + dry*dcy + drz*dcz;
  float tnd  = (float)(pr - pc) * (1.0f / 128.0f);

  float* o = Eo + (size_t)e * 5;
  o[0] = tnd;
  o[1] = d;
  o[2] = corr;
  o[3] = xr.y - xc.y;
  o[4] = xr.z - xc.z;
}

extern "C" void kernel_launch(void* const* d_in, const int* in_sizes, int n_in,
                              void* d_out, int out_size, void* d_ws, size_t ws_size,
                              hipStream_t stream) {
  const float* P  = (const float*)d_in[0];
  // d_in[1] = B (implied by structure: B[i] = i >> 7), unused
  const int* row  = (const int*)d_in[2];
  const int* col  = (const int*)d_in[3];
  const int N  = in_sizes[1];
  const int NE = in_sizes[2];
  const int G  = N / TRAJ;

  float* X      = (float*)d_out;                 // [N,4]
  float* Eo     = X + (size_t)N * 4;             // [NE,5]
  float* scales = Eo + (size_t)NE * 5;           // [G,2]
  float* u      = scales + (size_t)G * 2;        // [G,3]

  float* sf_ws   = (float*)d_ws;                 // [G]
  float* gmax_ws = sf_ws + G;                    // [G]
  float* fv_ws   = gmax_ws + G;                  // [G]
  float* pm_ws   = fv_ws + G;                    // [G]

  traj_node_kernel<<<G, 32, 0, stream>>>(P, X, scales, u, sf_ws, gmax_ws, fv_ws);

  int chunk = (G + 1023) / 1024;
  scan_max_kernel<<<1, 1024, 0, stream>>>(gmax_ws, pm_ws, G, chunk);

  finalize_cmx_kernel<<<(N + 255) / 256, 256, 0, stream>>>(X, pm_ws, fv_ws, N);

  traj_edge_kernel<<<(NE + 255) / 256, 256, 0, stream>>>(P, row, col, X, sf_ws, Eo, NE);
}